// LSTM_79826262163979
// MI455X (gfx1250) — compile-verified
//
#include <hip/hip_runtime.h>

// Problem constants (from reference): B=100000, T=8, I=32, H=10
#define BB   100000
#define TT   8
#define II   32
#define HH   10
#define GG   40      // 4*H gates
#define GP   48      // gates padded to 3 WMMA M-tiles of 16
#define NB   16      // batch elements per block
#define NCOL (NB*TT) // 128 (b,t) columns per block
#define NTHR 128     // 4 wave32 waves
#define FF   17      // MLP hidden width

typedef __attribute__((ext_vector_type(2))) float v2f;
typedef __attribute__((ext_vector_type(8))) float v8f;

__launch_bounds__(NTHR)
__global__ void lstm_fused_kernel(const float* __restrict__ x,
                                  const float* __restrict__ W_ih,
                                  const float* __restrict__ W_hh,
                                  const float* __restrict__ b_ih,
                                  const float* __restrict__ b_hh,
                                  const float* __restrict__ W1,
                                  const float* __restrict__ b1,
                                  const float* __restrict__ W2,
                                  const float* __restrict__ b2,
                                  float* __restrict__ out)
{
    // ---- LDS (≈47.5 KB total, well under the 320 KB WGP pool) ----
    __shared__ float wih_s[GP * II];      // 48x32 A operand, rows >=40 zeroed
    __shared__ float gxs[GP * NCOL];      // 48x128 gate pre-activations (WMMA output)
    __shared__ float bsum[GP];            // b_ih + b_hh (padded)
    __shared__ float whh_s[GG * HH];      // 40x10
    __shared__ float w1_s[FF * TT * HH];  // 17x80
    __shared__ float b1_s[FF];
    __shared__ float w2_s[FF];
    __shared__ float b2_s;
    __shared__ float hbuf[NB * HH];       // current h per element
    __shared__ float act_s[NB * GG];      // activated gates per step
    __shared__ float feat[NB * TT * HH];  // hs -> (16, 80) features
    __shared__ float hid[NB * FF];        // MLP hidden

    const int tid  = threadIdx.x;
    const int lane = tid & 31;
    const int wave = tid >> 5;
    const int b0   = blockIdx.x * NB;

    // ---- stage weights into LDS ----
    for (int i = tid; i < GP * II; i += NTHR) {
        int r = i >> 5;                       // row (gate) index
        wih_s[i] = (r < GG) ? W_ih[i] : 0.f;  // zero-pad rows 40..47
    }
    for (int i = tid; i < GG * HH; i += NTHR) whh_s[i] = W_hh[i];
    for (int i = tid; i < FF * TT * HH; i += NTHR) w1_s[i] = W1[i];
    if (tid < GP) bsum[tid] = (tid < GG) ? (b_ih[tid] + b_hh[tid]) : 0.f;
    if (tid < FF) { b1_s[tid] = b1[tid]; w2_s[tid] = W2[tid]; }
    if (tid == 0) b2_s = b2[0];
    for (int i = tid; i < NB * HH; i += NTHR) hbuf[i] = 0.f;
    __syncthreads();

    // =====================================================================
    // Phase 1: gxs[g][col] = sum_k W_ih[g][k] * x[b, t*32+k] + (b_ih+b_hh)[g]
    // via V_WMMA_F32_16X16X4_F32. Each wave handles 2 column-tiles of 16.
    // Columns: col = e*8 + t  (e = local batch element, t = timestep).
    // =====================================================================
    const int m16   = lane & 15;           // A: M row / B,D: N column within tile
    const int khalf = (lane >> 4) * 2;     // lane-half selects K pair {0,1} vs {2,3}
    for (int ct = 0; ct < 2; ++ct) {
        const int col = (wave * 2 + ct) * 16 + m16;     // local column 0..127
        const int bg  = b0 + (col >> 3);                // global batch element
        const int t   = col & 7;                        // timestep
        const float* xcol = x + (size_t)bg * (TT * II) + t * II; // 32 K values
        __builtin_prefetch(xcol, 0, 1);                 // global_prefetch path

        v8f acc0 = {}; v8f acc1 = {}; v8f acc2 = {};
#pragma unroll
        for (int kc = 0; kc < 8; ++kc) {                // K = 32 in chunks of 4
            const int k0 = kc * 4 + khalf;
            // B fragment (4x16 f32): lane holds X^T[k0..k0+1][col] = xcol[k0..k0+1]
            v2f bfrag = *(const v2f*)(xcol + k0);
            // A fragments (16x4 f32) for the three M-tiles, from LDS
            v2f a0 = *(const v2f*)(&wih_s[(m16 +  0) * II + k0]);
            v2f a1 = *(const v2f*)(&wih_s[(m16 + 16) * II + k0]);
            v2f a2 = *(const v2f*)(&wih_s[(m16 + 32) * II + k0]);
            acc0 = __builtin_amdgcn_wmma_f32_16x16x4_f32(false, a0, false, bfrag,
                                                         (short)0, acc0, false, false);
            acc1 = __builtin_amdgcn_wmma_f32_16x16x4_f32(false, a1, false, bfrag,
                                                         (short)0, acc1, false, false);
            acc2 = __builtin_amdgcn_wmma_f32_16x16x4_f32(false, a2, false, bfrag,
                                                         (short)0, acc2, false, false);
        }
        // D layout: VGPR r, lanes 0-15 -> M=r, lanes 16-31 -> M=8+r; N = lane&15
        const int rowoff = 8 * (lane >> 4);
#pragma unroll
        for (int r = 0; r < 8; ++r) {
            int g0 = r + rowoff;
            gxs[(g0     ) * NCOL + col] = acc0[r] + bsum[g0];
            gxs[(g0 + 16) * NCOL + col] = acc1[r] + bsum[g0 + 16];
            gxs[(g0 + 32) * NCOL + col] = acc2[r] + bsum[g0 + 32];
        }
    }
    __syncthreads();

    // =====================================================================
    // Phase 2: LSTM recurrence. 16 elements x 8 lanes; each lane computes 5
    // gates, then updates hidden units hu=s (and hu=s+8 for s<2).
    // =====================================================================
    const int e = tid >> 3;   // element 0..15
    const int s = tid & 7;    // sub-lane 0..7
    float c0 = 0.f, c1 = 0.f;
    const int hu0 = s, hu1 = s + 8;

    for (int t = 0; t < TT; ++t) {
#pragma unroll
        for (int q = 0; q < 5; ++q) {
            const int g = s * 5 + q;
            float a = gxs[g * NCOL + e * TT + t];
#pragma unroll
            for (int h = 0; h < HH; ++h)
                a += whh_s[g * HH + h] * hbuf[e * HH + h];
            float v;
            if (g >= 20 && g < 30) v = tanhf(a);                // cell gate
            else                   v = 1.f / (1.f + __expf(-a)); // i, f, o
            act_s[e * GG + g] = v;
        }
        __syncthreads();
        {
            float ig = act_s[e * GG +       hu0];
            float fg = act_s[e * GG + 10 +  hu0];
            float gg = act_s[e * GG + 20 +  hu0];
            float og = act_s[e * GG + 30 +  hu0];
            c0 = fg * c0 + ig * gg;
            float hv = og * tanhf(c0);
            hbuf[e * HH + hu0] = hv;
            feat[e * (TT * HH) + t * HH + hu0] = hv;
        }
        if (s < 2) {
            float ig = act_s[e * GG +       hu1];
            float fg = act_s[e * GG + 10 +  hu1];
            float gg = act_s[e * GG + 20 +  hu1];
            float og = act_s[e * GG + 30 +  hu1];
            c1 = fg * c1 + ig * gg;
            float hv = og * tanhf(c1);
            hbuf[e * HH + hu1] = hv;
            feat[e * (TT * HH) + t * HH + hu1] = hv;
        }
        __syncthreads();
    }

    // =====================================================================
    // Phase 3: MLP head. hid = sigmoid(feat @ W1^T + b1); out = hid @ W2^T + b2
    // =====================================================================
#pragma unroll
    for (int oi = 0; oi < 3; ++oi) {
        const int o = (oi == 0) ? s : (oi == 1) ? (s + 8) : 16;
        if (oi < 2 || s == 0) {
            float a = b1_s[o];
#pragma unroll
            for (int j = 0; j < TT * HH; ++j)
                a += feat[e * (TT * HH) + j] * w1_s[o * (TT * HH) + j];
            hid[e * FF + o] = 1.f / (1.f + __expf(-a));
        }
    }
    __syncthreads();
    if (s == 0) {
        float a = b2_s;
#pragma unroll
        for (int o = 0; o < FF; ++o) a += hid[e * FF + o] * w2_s[o];
        out[b0 + e] = a;
    }
}

extern "C" void kernel_launch(void* const* d_in, const int* in_sizes, int n_in,
                              void* d_out, int out_size, void* d_ws, size_t ws_size,
                              hipStream_t stream) {
    (void)in_sizes; (void)n_in; (void)d_ws; (void)ws_size; (void)out_size;
    const float* x    = (const float*)d_in[0];
    const float* W_ih = (const float*)d_in[1];
    const float* W_hh = (const float*)d_in[2];
    const float* b_ih = (const float*)d_in[3];
    const float* b_hh = (const float*)d_in[4];
    const float* W1   = (const float*)d_in[5];
    const float* b1   = (const float*)d_in[6];
    const float* W2   = (const float*)d_in[7];
    const float* b2   = (const float*)d_in[8];
    float* out = (float*)d_out;

    dim3 grid(BB / NB);   // 6250 blocks of 128 threads (4 waves each)
    lstm_fused_kernel<<<grid, NTHR, 0, stream>>>(x, W_ih, W_hh, b_ih, b_hh,
                                                 W1, b1, W2, b2, out);
}